// SATLayer_24223615550147
// MI455X (gfx1250) — compile-verified
//
#include <hip/hip_runtime.h>
#include <hip/hip_bf16.h>

#define N_NODES 100000
#define N_EDGES 1600000
#define D_IN    256
#define D_OUT   128

typedef __attribute__((ext_vector_type(2))) float v2f;
typedef __attribute__((ext_vector_type(8))) float v8f;

// ---------------------------------------------------------------------------
// Pass 0: init accumulators (harness poisons d_out / d_ws; re-init every call)
// ---------------------------------------------------------------------------
__global__ __launch_bounds__(256) void init_kernel(float* __restrict__ out,
                                                   float* __restrict__ smax,
                                                   float* __restrict__ denom) {
    const long long total = (long long)N_NODES * D_OUT;
    for (long long i = blockIdx.x * (long long)blockDim.x + threadIdx.x;
         i < total; i += (long long)gridDim.x * blockDim.x) {
        out[i] = 0.0f;
        if (i < N_NODES) {
            smax[i]  = -__builtin_inff();
            denom[i] = 0.0f;
        }
    }
}

// ---------------------------------------------------------------------------
// Pass 1: h = X @ W + b via V_WMMA_F32_16X16X4_F32
//   block = 256 threads (8 waves). blockIdx.x = M-tile (16 rows).
//   wave w computes the 16x16 tile at columns [16w, 16w+16).
// A layout (f32 16x4):  lane l -> M = l%16, K = 2*(l/16) + vgpr
// B layout (f32 4x16):  lane l -> N = l%16, K = 2*(l/16) + vgpr
// C/D layout:           lane l, vgpr v -> N = l%16, M = v + 8*(l/16)
// ---------------------------------------------------------------------------
__global__ __launch_bounds__(256) void gemm_kernel(const float* __restrict__ X,
                                                   const float* __restrict__ W,
                                                   const float* __restrict__ bias,
                                                   float* __restrict__ H) {
    const int wave  = threadIdx.x >> 5;      // 0..7 -> N tile
    const int lane  = threadIdx.x & 31;
    const int l16   = lane & 15;
    const int khalf = lane >> 4;             // 0 or 1

    const int mrow = blockIdx.x * 16 + l16;          // A row for this lane
    const int ncol = wave * 16 + l16;                // B/D column for this lane
    const float* __restrict__ xrow = X + (size_t)mrow * D_IN;

    v8f acc = {};
    #pragma unroll 4
    for (int k = 0; k < D_IN; k += 4) {
        const int ka = k + 2 * khalf;
        v2f a = *(const v2f*)(xrow + ka);            // K = ka, ka+1
        v2f bm;
        bm.x = W[(size_t)(ka    ) * D_OUT + ncol];
        bm.y = W[(size_t)(ka + 1) * D_OUT + ncol];
        acc = __builtin_amdgcn_wmma_f32_16x16x4_f32(
            /*neg_a=*/false, a, /*neg_b=*/false, bm,
            /*c_mod=*/(short)0, acc, /*reuse_a=*/false, /*reuse_b=*/false);
    }

    const float bv    = bias[ncol];
    const int   mbase = blockIdx.x * 16 + 8 * khalf; // rows mbase..mbase+7
    #pragma unroll
    for (int v = 0; v < 8; ++v) {
        H[(size_t)(mbase + v) * D_OUT + ncol] = acc[v] + bv;
    }
}

// ---------------------------------------------------------------------------
// Pass 2: a1[i] = h[i].wa1 + ba1 ; a2[i] = h[i].wa2 + ba2  (one wave per node)
// ---------------------------------------------------------------------------
__global__ __launch_bounds__(256) void proj_kernel(const float* __restrict__ H,
                                                   const float* __restrict__ wa1,
                                                   const float* __restrict__ ba1,
                                                   const float* __restrict__ wa2,
                                                   const float* __restrict__ ba2,
                                                   float* __restrict__ a1,
                                                   float* __restrict__ a2) {
    const int node = (int)((blockIdx.x * (long long)blockDim.x + threadIdx.x) >> 5);
    const int lane = threadIdx.x & 31;
    if (node >= N_NODES) return;
    const float* __restrict__ hrow = H + (size_t)node * D_OUT;
    float s1 = 0.0f, s2 = 0.0f;
    #pragma unroll
    for (int d = lane; d < D_OUT; d += 32) {
        const float hv = hrow[d];
        s1 += hv * wa1[d];
        s2 += hv * wa2[d];
    }
    #pragma unroll
    for (int off = 16; off > 0; off >>= 1) {
        s1 += __shfl_xor(s1, off, 32);
        s2 += __shfl_xor(s2, off, 32);
    }
    if (lane == 0) {
        a1[node] = s1 + ba1[0];
        a2[node] = s2 + ba2[0];
    }
}

// float atomic max via signed/unsigned bit trick (smax initialized to -inf)
__device__ __forceinline__ void atomic_max_float(float* addr, float val) {
    if (val >= 0.0f) atomicMax((int*)addr, __float_as_int(val));
    else             atomicMin((unsigned int*)addr, __float_as_uint(val));
}

// ---------------------------------------------------------------------------
// Pass 3: s = leaky_relu(a1[row]+a2[col]); segment max into smax[row]
// ---------------------------------------------------------------------------
__global__ __launch_bounds__(256) void edge_score_kernel(const int* __restrict__ row,
                                                         const int* __restrict__ col,
                                                         const float* __restrict__ a1,
                                                         const float* __restrict__ a2,
                                                         float* __restrict__ s,
                                                         float* __restrict__ smax) {
    const int e = (int)(blockIdx.x * (long long)blockDim.x + threadIdx.x);
    if (e >= N_EDGES) return;
    const int r = row[e];
    float v = a1[r] + a2[col[e]];
    v = (v > 0.0f) ? v : 0.01f * v;          // LeakyReLU slope 0.01
    s[e] = v;
    atomic_max_float(&smax[r], v);
}

// ---------------------------------------------------------------------------
// Pass 4: e = exp(s - smax[row]); denom[row] += e   (e overwrites s in place)
// ---------------------------------------------------------------------------
__global__ __launch_bounds__(256) void edge_exp_kernel(const int* __restrict__ row,
                                                       float* __restrict__ s,
                                                       const float* __restrict__ smax,
                                                       float* __restrict__ denom) {
    const int e = (int)(blockIdx.x * (long long)blockDim.x + threadIdx.x);
    if (e >= N_EDGES) return;
    const int r = row[e];
    const float ev = __expf(s[e] - smax[r]);
    s[e] = ev;
    atomicAdd(&denom[r], ev);
}

// ---------------------------------------------------------------------------
// Pass 5: out[row] += (e/denom[row]) * h[col]   (one wave per edge; 512B rows,
// gathers stay in L2 since h=51MB and out=51MB both fit the 192MB L2)
// ---------------------------------------------------------------------------
__global__ __launch_bounds__(256) void scatter_kernel(const int* __restrict__ row,
                                                      const int* __restrict__ col,
                                                      const float* __restrict__ eb,
                                                      const float* __restrict__ denom,
                                                      const float* __restrict__ H,
                                                      float* __restrict__ out) {
    const int e    = (int)((blockIdx.x * (long long)blockDim.x + threadIdx.x) >> 5);
    const int lane = threadIdx.x & 31;
    if (e >= N_EDGES) return;
    const int r = row[e];
    const int c = col[e];
    const float att = eb[e] / denom[r];
    const float4 hv = ((const float4*)(H + (size_t)c * D_OUT))[lane];
    float* orow = out + (size_t)r * D_OUT + lane * 4;
    atomicAdd(orow + 0, att * hv.x);
    atomicAdd(orow + 1, att * hv.y);
    atomicAdd(orow + 2, att * hv.z);
    atomicAdd(orow + 3, att * hv.w);
}

// ---------------------------------------------------------------------------
extern "C" void kernel_launch(void* const* d_in, const int* in_sizes, int n_in,
                              void* d_out, int out_size, void* d_ws, size_t ws_size,
                              hipStream_t stream) {
    const float* features = (const float*)d_in[0];
    const int*   edge_idx = (const int*)  d_in[1];   // [2, E] flat
    const float* W        = (const float*)d_in[2];
    const float* bvec     = (const float*)d_in[3];
    const float* wa1      = (const float*)d_in[4];
    const float* ba1      = (const float*)d_in[5];
    const float* wa2      = (const float*)d_in[6];
    const float* ba2      = (const float*)d_in[7];
    float* out = (float*)d_out;

    const int* row = edge_idx;
    const int* col = edge_idx + N_EDGES;

    // workspace layout
    float* H     = (float*)d_ws;                    // N_NODES * D_OUT
    float* a1    = H     + (size_t)N_NODES * D_OUT; // N_NODES
    float* a2    = a1    + N_NODES;                 // N_NODES
    float* smax  = a2    + N_NODES;                 // N_NODES
    float* denom = smax  + N_NODES;                 // N_NODES
    float* ebuf  = denom + N_NODES;                 // N_EDGES (s, then e)

    // Pass 0: init
    init_kernel<<<4096, 256, 0, stream>>>(out, smax, denom);

    // Pass 1: WMMA GEMM, one M-tile (16 rows) per block, 8 waves cover N=128
    gemm_kernel<<<N_NODES / 16, 256, 0, stream>>>(features, W, bvec, H);

    // Pass 2: attention projections, 1 wave/node, 8 waves/block
    proj_kernel<<<(N_NODES + 7) / 8, 256, 0, stream>>>(H, wa1, ba1, wa2, ba2, a1, a2);

    // Pass 3: edge scores + segment max
    edge_score_kernel<<<(N_EDGES + 255) / 256, 256, 0, stream>>>(row, col, a1, a2, ebuf, smax);

    // Pass 4: exp + segment sum
    edge_exp_kernel<<<(N_EDGES + 255) / 256, 256, 0, stream>>>(row, ebuf, smax, denom);

    // Pass 5: weighted scatter, 1 wave/edge, 8 waves/block
    scatter_kernel<<<(N_EDGES + 7) / 8, 256, 0, stream>>>(row, col, ebuf, denom, H, out);
}